// GNNPolicy_73409581023621
// MI455X (gfx1250) — compile-verified
//
#include <hip/hip_runtime.h>
#include <hip/hip_bf16.h>

// ---------------------------------------------------------------------------
// GCN 2-layer + edge logits for MI455X (gfx1250, wave32).
// GEMMs: V_WMMA_F32_16X16X4_F32 (fp32 matrix path, reference precision) with
// LDS-staged A strips (coalesced b128 global loads, bank-conflict-free ds
// reads). Aggregations: L2-resident scatter-adds (192MB L2 holds the whole
// 25.6MB feature table).
// ---------------------------------------------------------------------------

typedef __attribute__((ext_vector_type(2))) float v2f;
typedef __attribute__((ext_vector_type(8))) float v8f;

#define HAS_WMMA_F32 __has_builtin(__builtin_amdgcn_wmma_f32_16x16x4_f32)

// ---------------- degree / norm kernels ----------------

__global__ void k_deg_init(float* __restrict__ deg, int n) {
    int i = blockIdx.x * blockDim.x + threadIdx.x;
    if (i < n) deg[i] = 1.0f;  // self-loop contributes 1 to every node's degree
}

__global__ void k_deg_edges(const long long* __restrict__ dst, float* __restrict__ deg,
                            long long E) {
    long long e = (long long)blockIdx.x * blockDim.x + threadIdx.x;
    if (e < E) atomicAdd(&deg[(int)dst[e]], 1.0f);
}

__global__ void k_dinv(float* __restrict__ deg, int n) {
    int i = blockIdx.x * blockDim.x + threadIdx.x;
    if (i < n) deg[i] = rsqrtf(deg[i]);  // deg >= 1 always (self loop)
}

// ---------------- WMMA f32 GEMM: C[M,N] = op(A[M,K]) * B[K,N] ----------------
// One workgroup of N/16 waves owns a 16-row strip of C. The 16xK A strip is
// staged once into LDS (row stride 132 floats: 16-lane column reads hit 16
// distinct banks; rows stay 16B aligned for float4 staging). Each wave then
// computes one 16x16 tile, K stepped by 4 with V_WMMA_F32_16X16X4_F32.
//
// WMMA layouts (wave32):
//   A 16x4 f32: lanes 0-15 = rows, VGPR pair = {K+0,K+1}; lanes 16-31 = {K+2,K+3}
//   B 4x16:     VGPR0 = rows K+0 (lanes 0-15) / K+2 (lanes 16-31); VGPR1 = K+1/K+3
//   C/D:        VGPR v -> row v (lanes 0-15), row v+8 (lanes 16-31), col = lane&15

template<int K, int N, bool RELU>
__global__ __launch_bounds__(256) void k_gemm_wmma(const float* __restrict__ A,
                                                   const float* __restrict__ B,
                                                   float* __restrict__ C, int M) {
    constexpr int LDS_STRIDE = 132;              // 128 + 4 floats pad
    __shared__ float sA[16 * LDS_STRIDE];        // 8448 B

    const int lane = threadIdx.x;                // 0..31
    const int half = lane >> 4;                  // 0 or 1
    const int l15  = lane & 15;
    const int tm   = blockIdx.x * 16;            // strip row base (block-uniform)
    const int tn   = threadIdx.y * 16;           // this wave's N tile
    if (tm >= M) return;                         // block-uniform -> EXEC all-1s

    // ---- stage A strip [16][K] into LDS, coalesced float4, ReLU fused ----
    constexpr int NT = 32 * (N / 16);            // threads per block
    const int tid = threadIdx.y * 32 + lane;
    #pragma unroll
    for (int idx = tid; idx < 16 * (K / 4); idx += NT) {
        const int r  = idx / (K / 4);
        const int c4 = idx % (K / 4);
        float4 v = *(const float4*)(A + (long long)(tm + r) * K + c4 * 4);
        if (RELU) {
            v.x = fmaxf(v.x, 0.0f); v.y = fmaxf(v.y, 0.0f);
            v.z = fmaxf(v.z, 0.0f); v.w = fmaxf(v.w, 0.0f);
        }
        *(float4*)(&sA[r * LDS_STRIDE + c4 * 4]) = v;
    }
    __syncthreads();

#if HAS_WMMA_F32
    v8f acc = {};
    #pragma unroll 4
    for (int k = 0; k < K; k += 4) {
        const int ka = k + 2 * half;
        v2f av = {sA[l15 * LDS_STRIDE + ka], sA[l15 * LDS_STRIDE + ka + 1]};
        v2f bv = {B[(long long)ka * N + tn + l15],
                  B[(long long)(ka + 1) * N + tn + l15]};
        acc = __builtin_amdgcn_wmma_f32_16x16x4_f32(
            /*neg_a=*/false, av, /*neg_b=*/false, bv,
            /*c_mod=*/(short)0, acc, /*reuse_a=*/false, /*reuse_b=*/false);
    }
    #pragma unroll
    for (int v = 0; v < 8; ++v)
        C[(long long)(tm + v + 8 * half) * N + tn + l15] = acc[v];
#else
    // Scalar fallback (same output mapping) — should not be needed on gfx1250.
    #pragma unroll
    for (int v = 0; v < 8; ++v) {
        const int row = v + 8 * half;
        float s = 0.0f;
        for (int k = 0; k < K; ++k)
            s += sA[row * LDS_STRIDE + k] * B[(long long)k * N + tn + l15];
        C[(long long)(tm + row) * N + tn + l15] = s;
    }
#endif
}

// ---------------- aggregation init: self-loop term + bias ----------------
// out[n][f] = h[n][f] * dinv[n]^2 + bias[f]

template<int F>
__global__ void k_agg_init(const float* __restrict__ h, const float* __restrict__ dinv,
                           const float* __restrict__ bias, float* __restrict__ out, int n) {
    long long gid = (long long)blockIdx.x * blockDim.x + threadIdx.x;
    long long total = (long long)n * F;
    if (gid >= total) return;
    int node = (int)(gid / F);
    int f = (int)(gid % F);
    float di = dinv[node];
    out[gid] = h[gid] * di * di + bias[f];
}

// ---------------- edge scatter: out[dst] += h[src] * dinv[s]*dinv[d] --------
// F/4 threads per edge, float4 loads, f32 atomics (resolved in L2).

template<int F>
__global__ void k_edge_scatter(const long long* __restrict__ src,
                               const long long* __restrict__ dst,
                               const float* __restrict__ h,
                               const float* __restrict__ dinv,
                               float* __restrict__ out, long long E) {
    constexpr int TPE = F / 4;  // power of two (32 or 16)
    long long gid = (long long)blockIdx.x * blockDim.x + threadIdx.x;
    long long e = gid / TPE;
    int f4 = (int)(gid & (TPE - 1));
    if (e >= E) return;
    int s = (int)src[e], d = (int)dst[e];
    float w = dinv[s] * dinv[d];
    const float4 hv = *(const float4*)(h + (long long)s * F + f4 * 4);
    float* o = out + (long long)d * F + f4 * 4;
    atomicAdd(o + 0, hv.x * w);
    atomicAdd(o + 1, hv.y * w);
    atomicAdd(o + 2, hv.z * w);
    atomicAdd(o + 3, hv.w * w);
}

// ---------------- per-edge dot product over 64 features ----------------
// One wave32 per edge, float2 per lane, shfl_xor butterfly reduction.

__global__ void k_edge_logits(const long long* __restrict__ src,
                              const long long* __restrict__ dst,
                              const float* __restrict__ h,
                              float* __restrict__ out, long long E) {
    int lane = threadIdx.x & 31;
    long long e = ((long long)blockIdx.x * blockDim.x + threadIdx.x) >> 5;
    if (e >= E) return;
    int s = (int)src[e], d = (int)dst[e];
    const float2 a = *(const float2*)(h + (long long)s * 64 + lane * 2);
    const float2 b = *(const float2*)(h + (long long)d * 64 + lane * 2);
    float p = a.x * b.x + a.y * b.y;
    #pragma unroll
    for (int off = 16; off > 0; off >>= 1) p += __shfl_xor(p, off, 32);
    if (lane == 0) out[e] = p;
}

// ---------------------------------------------------------------------------

static inline int divUp(long long a, long long b) { return (int)((a + b - 1) / b); }

extern "C" void kernel_launch(void* const* d_in, const int* in_sizes, int n_in,
                              void* d_out, int out_size, void* d_ws, size_t ws_size,
                              hipStream_t stream) {
    const float*     x   = (const float*)d_in[0];      // [N,128]
    const long long* ei  = (const long long*)d_in[1];  // [2,E] int64
    const float*     W1  = (const float*)d_in[2];      // [128,128]
    const float*     b1  = (const float*)d_in[3];      // [128]
    const float*     W2  = (const float*)d_in[4];      // [128,64]
    const float*     b2  = (const float*)d_in[5];      // [64]

    const int  N = in_sizes[0] / 128;          // 50000
    const long long E = in_sizes[1] / 2;       // 800000
    const long long EN = E;
    const long long* src = ei;
    const long long* dst = ei + E;

    // workspace layout: dinv [N] | bufA [N*128] | bufB [N*128]
    char* ws = (char*)d_ws;
    size_t offA = ((size_t)N * 4 + 255) & ~(size_t)255;
    size_t offB = offA + (size_t)N * 128 * 4;
    float* dinv = (float*)ws;
    float* bufA = (float*)(ws + offA);   // h1, later h2
    float* bufB = (float*)(ws + offB);   // agg1, later h_final
    float* logits = (float*)d_out;

    const int strips = (N + 15) / 16;    // 3125 (exact: 50000 = 3125*16)

    // ---- degrees & symmetric norm ----
    k_deg_init<<<divUp(N, 256), 256, 0, stream>>>(dinv, N);
    k_deg_edges<<<divUp(EN, 256), 256, 0, stream>>>(dst, dinv, EN);
    k_dinv<<<divUp(N, 256), 256, 0, stream>>>(dinv, N);

    // ---- layer 1: h1 = x @ W1 ----
    k_gemm_wmma<128, 128, false><<<dim3(strips), dim3(32, 8), 0, stream>>>(x, W1, bufA, N);
    // agg1 = h1*dinv^2 + b1 ; scatter edges
    k_agg_init<128><<<divUp((long long)N * 128, 256), 256, 0, stream>>>(bufA, dinv, b1, bufB, N);
    k_edge_scatter<128><<<divUp(EN * 32, 256), 256, 0, stream>>>(src, dst, bufA, dinv, bufB, EN);

    // ---- layer 2: h2 = relu(agg1) @ W2  (ReLU fused at LDS staging) ----
    k_gemm_wmma<128, 64, true><<<dim3(strips), dim3(32, 4), 0, stream>>>(bufB, W2, bufA, N);
    // h_final = h2*dinv^2 + b2 ; scatter edges (overwrites bufB, already consumed)
    k_agg_init<64><<<divUp((long long)N * 64, 256), 256, 0, stream>>>(bufA, dinv, b2, bufB, N);
    k_edge_scatter<64><<<divUp(EN * 16, 256), 256, 0, stream>>>(src, dst, bufA, dinv, bufB, EN);

    // ---- per-edge logits ----
    k_edge_logits<<<divUp(EN * 32, 256), 256, 0, stream>>>(src, dst, bufB, logits, EN);
}